// MyModel_68771016343836
// MI455X (gfx1250) — compile-verified
//
#include <hip/hip_runtime.h>

typedef __attribute__((ext_vector_type(2))) float v2f;
typedef __attribute__((ext_vector_type(8))) float v8f;
typedef __attribute__((ext_vector_type(4))) int   v4i;

#define D_DIM   640
#define NWAY    32
#define KSHOT   32
#define INV_TAU 10.0f

// GEMM tiling
#define CH      32            // K-chunk staged in LDS
#define LSTR    36            // padded LDS row stride (floats): 144 B, 16B aligned, bank-conflict-free
#define ROWS    128           // query rows per block (8 waves x 16)
#define NCHUNK  (D_DIM / CH)  // 20
#define KBL     (CH / 4)      // 8 WMMA K-steps per chunk
#define NKB     (D_DIM / 4)   // 160 K-blocks total

#if defined(__HIP_DEVICE_COMPILE__) && __has_builtin(__builtin_amdgcn_global_load_async_to_lds_b128)
#define HAVE_ASYNC_LDS 1
#else
#define HAVE_ASYNC_LDS 0
#endif

static __device__ __forceinline__ void wait_async0() {
#if HAVE_ASYNC_LDS
#if __has_builtin(__builtin_amdgcn_s_wait_asynccnt)
    __builtin_amdgcn_s_wait_asynccnt(0);
#else
    asm volatile("s_wait_asynccnt 0x0" ::: "memory");
#endif
#endif
}

// ---------------------------------------------------------------------------
// Kernel 1: prototypes + swizzle into WMMA B-fragment layout.
// wsB layout (float2): [tile t in 0..1][kb in 0..159][lane in 0..31]
//   lane n      (n<16): (proto[t*16+n][4kb+0], proto[t*16+n][4kb+1])  -> B VGPR0/1, K=0/1
//   lane n+16        : (proto[t*16+n][4kb+2], proto[t*16+n][4kb+3])  -> B VGPR0/1, K=2/3
// ---------------------------------------------------------------------------
__global__ __launch_bounds__(256) void proto_swizzle_kernel(
    const float* __restrict__ S, float2* __restrict__ wsB)
{
    __shared__ float Pi[D_DIM];
    __shared__ float red[8];
    __shared__ float dotsL[NWAY], snrm2L[NWAY], wL[NWAY];

    const int i    = blockIdx.x;
    const int tid  = threadIdx.x;
    const int lane = tid & 31;
    const int wave = tid >> 5;

    // P_i[d] = mean_k support[k*32 + i][d]  (faithful to the reference's strided view)
    float pn_part = 0.f;
    for (int d = tid; d < D_DIM; d += 256) {
        float acc = 0.f;
        #pragma unroll
        for (int k = 0; k < KSHOT; ++k)
            acc += S[(size_t)(k * NWAY + i) * D_DIM + d];
        acc *= (1.0f / KSHOT);
        Pi[d] = acc;
        pn_part = fmaf(acc, acc, pn_part);
    }
    #pragma unroll
    for (int off = 16; off; off >>= 1) pn_part += __shfl_down(pn_part, off, 32);
    if (lane == 0) red[wave] = pn_part;
    __syncthreads();

    // dots[j] = <P_i, S[i*32+j]>, snrm2[j] = |S[i*32+j]|^2 ; each wave does 4 shots
    for (int t = 0; t < 4; ++t) {
        const int j = wave * 4 + t;
        const float* row = S + (size_t)(i * KSHOT + j) * D_DIM;
        float ad = 0.f, an = 0.f;
        for (int d = lane; d < D_DIM; d += 32) {
            float s = row[d];
            ad = fmaf(Pi[d], s, ad);
            an = fmaf(s, s, an);
        }
        #pragma unroll
        for (int off = 16; off; off >>= 1) {
            ad += __shfl_down(ad, off, 32);
            an += __shfl_down(an, off, 32);
        }
        if (lane == 0) { dotsL[j] = ad; snrm2L[j] = an; }
    }
    __syncthreads();

    // one-wave softmax over cosine similarities
    if (wave == 0) {
        float pn2 = 0.f;
        #pragma unroll
        for (int r = 0; r < 8; ++r) pn2 += red[r];
        float denom = fmaxf(sqrtf(pn2) * sqrtf(snrm2L[lane]), 1e-8f);
        float cosv  = dotsL[lane] / denom;
        float m = cosv;
        #pragma unroll
        for (int off = 16; off; off >>= 1) m = fmaxf(m, __shfl_xor(m, off, 32));
        float e = expf(cosv - m);
        float ssum = e;
        #pragma unroll
        for (int off = 16; off; off >>= 1) ssum += __shfl_xor(ssum, off, 32);
        wL[lane] = e / ssum;
    }
    __syncthreads();

    // proto_i = sum_j w[j] * S[i*32+j], then L2 normalize
    float protod[3] = {0.f, 0.f, 0.f};
    float nrm_part = 0.f;
    {
        int r = 0;
        for (int d = tid; d < D_DIM; d += 256, ++r) {
            float acc = 0.f;
            #pragma unroll
            for (int j = 0; j < KSHOT; ++j)
                acc = fmaf(wL[j], S[(size_t)(i * KSHOT + j) * D_DIM + d], acc);
            protod[r] = acc;
            nrm_part  = fmaf(acc, acc, nrm_part);
        }
    }
    #pragma unroll
    for (int off = 16; off; off >>= 1) nrm_part += __shfl_down(nrm_part, off, 32);
    if (lane == 0) red[wave] = nrm_part;
    __syncthreads();

    float tot = 0.f;
    #pragma unroll
    for (int r = 0; r < 8; ++r) tot += red[r];
    const float scale = 1.0f / fmaxf(sqrtf(tot), 1e-12f);

    {
        int r = 0;
        for (int d = tid; d < D_DIM; d += 256, ++r) Pi[d] = protod[r] * scale;
    }
    __syncthreads();

    // emit swizzled B fragments: this row feeds column n of tile t
    if (tid < NKB) {
        const int t = i >> 4, n = i & 15;
        float2 v0, v1;
        v0.x = Pi[4 * tid + 0]; v0.y = Pi[4 * tid + 1];
        v1.x = Pi[4 * tid + 2]; v1.y = Pi[4 * tid + 3];
        wsB[((size_t)t * NKB + tid) * 32 + n]      = v0;
        wsB[((size_t)t * NKB + tid) * 32 + n + 16] = v1;
    }
}

// ---------------------------------------------------------------------------
// Kernel 2: logits = query @ proto^T * (1/tau) via V_WMMA_F32_16X16X4_F32.
// 8 waves/block, each wave owns a 16x32 output tile. Query K-chunks move
// memory->LDS with GLOBAL_LOAD_ASYNC_TO_LDS_B128 (no data VGPRs, ASYNCcnt),
// double-buffered so the DMA overlaps the 16 WMMAs of the current chunk.
// ---------------------------------------------------------------------------
__global__ __launch_bounds__(256) void gemm_wmma_kernel(
    const float* __restrict__ Q, const float* __restrict__ wsB,
    float* __restrict__ out)
{
    __shared__ float lds[2][ROWS * LSTR];   // 2 * 128*36*4 = 36864 B

    const int tid   = threadIdx.x;
    const int lane  = tid & 31;
    const int wave  = tid >> 5;
    const size_t row0 = (size_t)blockIdx.x * ROWS;

    // A-fragment addressing for 16x4 f32: lanes 0-15 -> K=0,1 ; lanes 16-31 -> K=2,3
    const int mrow  = wave * 16 + (lane & 15);
    const int khalf = (lane >> 4) * 2;

    v8f acc0 = {};
    v8f acc1 = {};

    // stage one K-chunk: 128 rows x 32 floats = 1024 float4s, 4 per thread
    auto issue_chunk = [&](int cc, int buf) {
        const float* src = Q + row0 * D_DIM + (size_t)cc * CH;
        #pragma unroll
        for (int r = 0; r < 4; ++r) {
            const int idx = tid + 256 * r;          // 0..1023
            const int rr  = idx >> 3;               // row 0..127 (8 float4/row)
            const int c4  = (idx & 7) << 2;         // float col 0..28
#if HAVE_ASYNC_LDS
            __builtin_amdgcn_global_load_async_to_lds_b128(
                (v4i*)(src + (size_t)rr * D_DIM + c4),
                (v4i*)&lds[buf][rr * LSTR + c4],
                0, 0);
#else
            *(float4*)&lds[buf][rr * LSTR + c4] =
                *(const float4*)(src + (size_t)rr * D_DIM + c4);
#endif
        }
    };

    issue_chunk(0, 0);                     // prologue

    for (int c = 0; c < NCHUNK; ++c) {
        wait_async0();                     // my share of chunk c has landed in LDS
        __syncthreads();                   // everyone's share visible
        if (c + 1 < NCHUNK)
            issue_chunk(c + 1, (c + 1) & 1);   // DMA next chunk during WMMAs

        const int buf = c & 1;
        const float* abase  = &lds[buf][mrow * LSTR + khalf];
        const float* b0base = wsB + ((size_t)(c * KBL) * 32 + lane) * 2;
        const float* b1base = b0base + (size_t)NKB * 32 * 2;
        #pragma unroll
        for (int kbl = 0; kbl < KBL; ++kbl) {
            v2f a  = *(const v2f*)(abase  + kbl * 4);
            v2f b0 = *(const v2f*)(b0base + kbl * 64);
            v2f b1 = *(const v2f*)(b1base + kbl * 64);
            acc0 = __builtin_amdgcn_wmma_f32_16x16x4_f32(
                       false, a, false, b0, (short)0, acc0, false, false);
            acc1 = __builtin_amdgcn_wmma_f32_16x16x4_f32(
                       false, a, false, b1, (short)0, acc1, false, false);
        }
    }

    // C/D layout: VGPR r -> M = r (lanes 0-15) / r+8 (lanes 16-31), N = lane&15
    const size_t orow0 = row0 + (size_t)wave * 16 + (size_t)(lane >> 4) * 8;
    const int nn = lane & 15;
    #pragma unroll
    for (int r = 0; r < 8; ++r) {
        const size_t o = (orow0 + r) * NWAY + nn;
        out[o]      = acc0[r] * INV_TAU;
        out[o + 16] = acc1[r] * INV_TAU;
    }
}

// ---------------------------------------------------------------------------
extern "C" void kernel_launch(void* const* d_in, const int* in_sizes, int n_in,
                              void* d_out, int out_size, void* d_ws, size_t ws_size,
                              hipStream_t stream)
{
    (void)n_in; (void)out_size; (void)ws_size;
    const float* support = (const float*)d_in[0];
    const float* query   = (const float*)d_in[1];
    float*       out     = (float*)d_out;
    float2*      wsB     = (float2*)d_ws;      // needs 2*160*32*8 = 81920 B

    proto_swizzle_kernel<<<NWAY, 256, 0, stream>>>(support, wsB);

    const int qrows   = in_sizes[1] / D_DIM;   // 131072
    const int nblocks = qrows / ROWS;          // 1024
    gemm_wmma_kernel<<<nblocks, 256, 0, stream>>>(query, (const float*)d_ws, out);
}